// NWNet_575525618150
// MI455X (gfx1250) — compile-verified
//
#include <hip/hip_runtime.h>
#include <math.h>

typedef __attribute__((ext_vector_type(2))) float v2f;
typedef __attribute__((ext_vector_type(8))) float v8f;

#define B_TOT   64
#define S_TOT   100000
#define D_DIM   256
#define C_TOT   1000
#define TS      32                 // s rows per block in score kernel
#define LDSS    (D_DIM + 4)        // padded LDS row stride (bank-conflict-free)
#define NBLK1   (S_TOT / TS)       // 3125
#define NCHUNK  32
#define CHUNK_S (S_TOT / NCHUNK)   // 3125
#define EPS_F   1e-12f

#if __has_builtin(__builtin_amdgcn_global_load_async_to_lds_b128) && \
    __has_builtin(__builtin_amdgcn_s_wait_asynccnt)
#define USE_ASYNC_LDS 1
typedef int v4i_vs __attribute__((vector_size(16)));
typedef __attribute__((address_space(1))) v4i_vs* gvec_t;   // global int4*
typedef __attribute__((address_space(3))) v4i_vs* lvec_t;   // LDS    int4*
#else
#define USE_ASYNC_LDS 0
#endif

// ---------------------------------------------------------------- q2 ----
__global__ __launch_bounds__(64) void q2_kernel(const float* __restrict__ qfeat,
                                                float* __restrict__ q2) {
    int b = threadIdx.x;
    if (b < B_TOT) {
        float s = 0.f;
        const float* r = qfeat + (size_t)b * D_DIM;
        #pragma unroll 4
        for (int k = 0; k < D_DIM; ++k) { float v = r[k]; s += v * v; }
        q2[b] = s;
    }
}

// ------------------------------------------------------------ scores ----
// Block: 256 threads = 8 waves. Waves tile [2 x s16] x [4 x b16].
// Wave w: mtile = w/4 (s sub-tile), ntile = w%4 (b tile).
__global__ __launch_bounds__(256) void score_kernel(const float* __restrict__ qfeat,
                                                    const float* __restrict__ sfeat,
                                                    const float* __restrict__ q2,
                                                    float* __restrict__ scores,
                                                    float* __restrict__ pmax) {
    __shared__ float lds_s[TS * LDSS];     // 32 x 260 floats = 33.3 KB
    __shared__ float s2[TS];
    __shared__ float wmax[8 * 16];

    const int tid = threadIdx.x;
    const int s0  = blockIdx.x * TS;

    // Stage the 32x256 sfeat tile in LDS.
#if USE_ASYNC_LDS
    // Direct memory->LDS async copy (ASYNCcnt-tracked), no VGPR round trip.
    for (int idx = tid; idx < TS * (D_DIM / 4); idx += 256) {
        int row = idx >> 6;            // /64
        int c4  = idx & 63;            // %64
        const float* gsrc = sfeat + (size_t)(s0 + row) * D_DIM + c4 * 4;
        __builtin_amdgcn_global_load_async_to_lds_b128(
            (gvec_t)gsrc, (lvec_t)&lds_s[row * LDSS + c4 * 4], 0, 0);
        if (s0 + TS < S_TOT)
            __builtin_prefetch(sfeat + (size_t)(s0 + TS + row) * D_DIM + c4 * 4, 0, 1);
    }
    __builtin_amdgcn_s_wait_asynccnt(0);
    __syncthreads();
#else
    for (int idx = tid; idx < TS * (D_DIM / 4); idx += 256) {
        int row = idx >> 6;
        int c4  = idx & 63;
        float4 v = reinterpret_cast<const float4*>(
                       sfeat + (size_t)(s0 + row) * D_DIM)[c4];
        *reinterpret_cast<float4*>(&lds_s[row * LDSS + c4 * 4]) = v;
        if (s0 + TS < S_TOT)
            __builtin_prefetch(sfeat + (size_t)(s0 + TS + row) * D_DIM + c4 * 4, 0, 1);
    }
    __syncthreads();
#endif

    // Per-row squared norms of the staged tile.
    if (tid < TS) {
        float acc = 0.f;
        const float* r = &lds_s[tid * LDSS];
        #pragma unroll 4
        for (int k = 0; k < D_DIM; ++k) acc += r[k] * r[k];
        s2[tid] = acc;
    }
    __syncthreads();

    const int wave  = tid >> 5;
    const int lane  = tid & 31;
    const int mtile = wave >> 2;     // 0..1
    const int ntile = wave & 3;      // 0..3
    const int lhalf = lane >> 4;     // 0..1
    const int l16   = lane & 15;

    const float* arow = &lds_s[(mtile * 16 + l16) * LDSS];
    const float* brow = qfeat + (size_t)(ntile * 16 + l16) * D_DIM;

    // D = A(16x4, s-major) * B(4x16, q) + C, chained over K=256.
    // Two independent accumulator chains to break WMMA->WMMA latency.
    v8f acc0 = {}, acc1 = {};
    #pragma unroll 4
    for (int k0 = 0; k0 < D_DIM; k0 += 8) {
        int kk = k0 + 2 * lhalf;
        float2 af0 = *reinterpret_cast<const float2*>(arow + kk);
        float2 bf0 = *reinterpret_cast<const float2*>(brow + kk);
        float2 af1 = *reinterpret_cast<const float2*>(arow + kk + 4);
        float2 bf1 = *reinterpret_cast<const float2*>(brow + kk + 4);
        v2f a0; a0.x = af0.x; a0.y = af0.y;
        v2f b0; b0.x = bf0.x; b0.y = bf0.y;
        v2f a1; a1.x = af1.x; a1.y = af1.y;
        v2f b1; b1.x = bf1.x; b1.y = bf1.y;
        acc0 = __builtin_amdgcn_wmma_f32_16x16x4_f32(
                   false, a0, false, b0, (short)0, acc0, false, false);
        acc1 = __builtin_amdgcn_wmma_f32_16x16x4_f32(
                   false, a1, false, b1, (short)0, acc1, false, false);
    }
    v8f acc = acc0 + acc1;

    // Epilogue: d2 -> score, store, per-b running max.
    const int   b_idx = ntile * 16 + l16;
    const float qq    = q2[b_idx];
    float lmax = -INFINITY;
    #pragma unroll
    for (int r = 0; r < 8; ++r) {
        int   s_local = mtile * 16 + 8 * lhalf + r;   // D layout: M = r + 8*(lane>=16)
        float d2 = qq + s2[s_local] - 2.0f * acc[r];
        float sc = -sqrtf(fmaxf(d2, EPS_F));
        scores[(size_t)(s0 + s_local) * B_TOT + b_idx] = sc;
        lmax = fmaxf(lmax, sc);
    }
    lmax = fmaxf(lmax, __shfl_xor(lmax, 16, 32));     // merge lane halves (same b)
    if (lhalf == 0) wmax[wave * 16 + l16] = lmax;
    __syncthreads();
    if (tid < B_TOT) {
        int nt = tid >> 4, bl = tid & 15;
        float m = fmaxf(wmax[nt * 16 + bl], wmax[(4 + nt) * 16 + bl]);
        pmax[(size_t)blockIdx.x * B_TOT + tid] = m;
    }
}

// --------------------------------------------------------- max reduce ----
__global__ __launch_bounds__(256) void reduce_max_kernel(const float* __restrict__ pmax,
                                                         float* __restrict__ gmax,
                                                         int nblocks) {
    __shared__ float sm[256];
    int t = threadIdx.x, b = t & 63, j = t >> 6;
    float m = -INFINITY;
    for (int i = j; i < nblocks; i += 4)
        m = fmaxf(m, pmax[(size_t)i * B_TOT + b]);
    sm[t] = m;
    __syncthreads();
    if (t < B_TOT)
        gmax[t] = fmaxf(fmaxf(sm[b], sm[64 + b]), fmaxf(sm[128 + b], sm[192 + b]));
}

// ---------------------------------------------- class-binned exp sums ----
// grid: (NCHUNK, 4).  Block owns s-range [chunk*CHUNK_S, ...) x 16 b's.
__global__ __launch_bounds__(256) void bin_kernel(const float* __restrict__ scores,
                                                  const int* __restrict__ sy,
                                                  const float* __restrict__ gmax,
                                                  float* __restrict__ U_part) {
    __shared__ float hist[16 * C_TOT];   // 62.5 KB
    const int t     = threadIdx.x;
    const int chunk = blockIdx.x;
    const int bg    = blockIdx.y;

    for (int i = t; i < 16 * C_TOT; i += 256) hist[i] = 0.f;
    float m[16];
    #pragma unroll
    for (int i = 0; i < 16; ++i) m[i] = gmax[bg * 16 + i];
    __syncthreads();

    const int s_begin = chunk * CHUNK_S;
    for (int s = s_begin + t; s < s_begin + CHUNK_S; s += 256) {
        int c = sy[s];
        const float4* row4 = reinterpret_cast<const float4*>(
                                 scores + (size_t)s * B_TOT + bg * 16);
        float4 a0 = row4[0], a1 = row4[1], a2 = row4[2], a3 = row4[3];
        float sc[16] = { a0.x, a0.y, a0.z, a0.w,  a1.x, a1.y, a1.z, a1.w,
                         a2.x, a2.y, a2.z, a2.w,  a3.x, a3.y, a3.z, a3.w };
        #pragma unroll
        for (int i = 0; i < 16; ++i)
            atomicAdd(&hist[i * C_TOT + c], __expf(sc[i] - m[i]));
    }
    __syncthreads();

    for (int i = t; i < 16 * C_TOT; i += 256) {
        int bl = i / C_TOT, c = i % C_TOT;
        U_part[((size_t)chunk * B_TOT + (bg * 16 + bl)) * C_TOT + c] = hist[i];
    }
}

// ---------------------------------------------------- chunk reduction ----
__global__ __launch_bounds__(256) void reduce_U_kernel(const float* __restrict__ U_part,
                                                       float* __restrict__ Ufinal) {
    int idx = blockIdx.x * 256 + threadIdx.x;
    if (idx < B_TOT * C_TOT) {
        float s = 0.f;
        #pragma unroll 4
        for (int ch = 0; ch < NCHUNK; ++ch)
            s += U_part[(size_t)ch * B_TOT * C_TOT + idx];
        Ufinal[idx] = s;
    }
}

// -------------------------------------------- Z = row-sum, reciprocal ----
__global__ __launch_bounds__(256) void zrecip_kernel(const float* __restrict__ Ufinal,
                                                     float* __restrict__ rz) {
    __shared__ float sm[256];
    int t = threadIdx.x, b = t & 63, j = t >> 6;
    float s = 0.f;
    for (int c = j; c < C_TOT; c += 4) s += Ufinal[(size_t)b * C_TOT + c];
    sm[t] = s;
    __syncthreads();
    if (t < B_TOT)
        rz[t] = 1.0f / (sm[b] + sm[64 + b] + sm[128 + b] + sm[192 + b]);
}

// ----------------------------------------------------------- finalize ----
__global__ __launch_bounds__(256) void final_kernel(const float* __restrict__ Ufinal,
                                                    const float* __restrict__ rz,
                                                    float* __restrict__ out) {
    int idx = blockIdx.x * 256 + threadIdx.x;
    if (idx < B_TOT * C_TOT)
        out[idx] = __expf(Ufinal[idx] * rz[idx / C_TOT]);
}

// ------------------------------------------------------------- launch ----
extern "C" void kernel_launch(void* const* d_in, const int* in_sizes, int n_in,
                              void* d_out, int out_size, void* d_ws, size_t ws_size,
                              hipStream_t stream) {
    const float* qfeat = (const float*)d_in[0];
    const float* sfeat = (const float*)d_in[1];
    const int*   sy    = (const int*)d_in[2];
    (void)in_sizes; (void)n_in; (void)out_size; (void)ws_size;

    // Workspace carve-up (floats).
    float* w      = (float*)d_ws;
    float* q2     = w;                                   //     64
    float* gmax   = w + 64;                              //     64
    float* rz     = w + 128;                             //     64
    float* pmax   = w + 256;                             // 3125*64 = 200000
    float* Ufinal = pmax + (size_t)NBLK1 * B_TOT;        //  64000
    float* U_part = Ufinal + (size_t)B_TOT * C_TOT;      // 32*64000 = 2048000
    float* scores = U_part + (size_t)NCHUNK * B_TOT * C_TOT;  // 100000*64

    q2_kernel<<<1, 64, 0, stream>>>(qfeat, q2);
    score_kernel<<<NBLK1, 256, 0, stream>>>(qfeat, sfeat, q2, scores, pmax);
    reduce_max_kernel<<<1, 256, 0, stream>>>(pmax, gmax, NBLK1);
    bin_kernel<<<dim3(NCHUNK, 4, 1), 256, 0, stream>>>(scores, sy, gmax, U_part);
    reduce_U_kernel<<<(B_TOT * C_TOT + 255) / 256, 256, 0, stream>>>(U_part, Ufinal);
    zrecip_kernel<<<1, 256, 0, stream>>>(Ufinal, rz);
    final_kernel<<<(B_TOT * C_TOT + 255) / 256, 256, 0, stream>>>(Ufinal, rz,
                                                                  (float*)d_out);
}